// LSTM_Peephole_build_51951924412948
// MI455X (gfx1250) — compile-verified
//
#include <hip/hip_runtime.h>
#include <math.h>

// ---------------------------------------------------------------------------
// Peephole LSTM step, fp32, WMMA f32 16x16x4 (exact fp32 math like reference).
//   K1: gates  -> Ct (out), zo (ws)
//   K2: Ot = sigmoid(zo + Ct@w_oc); Ht = Ot*tanh(Ct)
//   K3: Yt = sigmoid(Ht@w_op)
// Output layout: [Yt (4096x32000) | Ht (4096x1024) | Ct (4096x1024)]
// Double-buffered LDS staging via GLOBAL_LOAD_ASYNC_TO_LDS_B128 (inline asm,
// ASYNCcnt-tracked) on gfx1250; sync float4 copies otherwise.
// ---------------------------------------------------------------------------

#define BM 64
#define BN 64
#define KT 16
#define AP 20   // A tile LDS pitch in floats (16 + 4 pad -> 80B rows, 16B aligned)
#define BP 68   // B tile LDS pitch in floats (64 + 4 pad -> 272B rows, 16B aligned)

typedef __attribute__((ext_vector_type(2))) float v2f;
typedef __attribute__((ext_vector_type(8))) float v8f;

#if defined(__gfx1250__)
#define HAVE_ASYNC_LDS 1
#else
#define HAVE_ASYNC_LDS 0
#endif

static __device__ __forceinline__ v8f wmma_f32x4(v2f a, v2f b, v8f c) {
  // (neg_a, A, neg_b, B, c_mod, C, reuse_a, reuse_b) -> v_wmma_f32_16x16x4_f32
  return __builtin_amdgcn_wmma_f32_16x16x4_f32(false, a, false, b, (short)0, c,
                                               false, false);
}

static __device__ __forceinline__ float sigmoidf_(float v) {
  return 1.0f / (1.0f + __expf(-v));
}

// One 16-byte global->LDS copy per calling thread.
// On AMDGPU a generic pointer to __shared__ data is aperture_hi<<32 | lds_addr
// (ISA 10.2: LDS aperture keeps addr[31:0]), so the low 32 bits are the LDS
// byte address the async instruction's VDST operand expects.
static __device__ __forceinline__ void async_copy16(const float* __restrict__ g,
                                                    float* __restrict__ l) {
#if HAVE_ASYNC_LDS
  const unsigned loff = (unsigned)(size_t)l;
  asm volatile("global_load_async_to_lds_b128 %0, %1, off"
               :
               : "v"(loff), "v"(g)
               : "memory");
#else
  *reinterpret_cast<float4*>(l) = *reinterpret_cast<const float4*>(g);
#endif
}

template <int N>
static __device__ __forceinline__ void wait_async() {
#if HAVE_ASYNC_LDS
  asm volatile("s_wait_asynccnt %0" : : "i"(N) : "memory");
#endif
}

// Cooperative tile staging: 256 threads move one 64x16 (A) or 16x64 (B)
// fp32 tile with one async b128 per thread.
static __device__ __forceinline__ void tile_A(float* __restrict__ dst,
                                              const float* __restrict__ src,
                                              int row0, int kk, int ld,
                                              int tid) {
  const int r = tid >> 2;          // 0..63
  const int q = (tid & 3) << 2;    // 0,4,8,12
  async_copy16(src + (size_t)(row0 + r) * ld + kk + q, dst + r * AP + q);
}

static __device__ __forceinline__ void tile_B(float* __restrict__ dst,
                                              const float* __restrict__ src,
                                              int kk, int col0, int ld,
                                              int tid) {
  const int kr = tid >> 4;         // 0..15
  const int nq = (tid & 15) << 2;  // 0..60
  async_copy16(src + (size_t)(kk + kr) * ld + col0 + nq, dst + kr * BP + nq);
}

// ISA layout: A 16x4 f32 -> lane holds row M=lane&15, K pair selected by lane>=16.
static __device__ __forceinline__ v2f a_frag(const float* __restrict__ As,
                                             int row0, int k, int lane) {
  const int m = lane & 15;
  const int kq = k + ((lane >> 4) << 1);
  const float* p = As + (row0 + m) * AP + kq;
  v2f r;
  r.x = p[0];
  r.y = p[1];
  return r;
}

// B 4x16 f32 -> lane holds col N=lane&15, K pair selected by lane>=16.
static __device__ __forceinline__ v2f b_frag(const float* __restrict__ Bs, int k,
                                             int col0, int lane) {
  const int n = lane & 15;
  const int kq = k + ((lane >> 4) << 1);
  const float* p = Bs + kq * BP + col0 + n;
  v2f r;
  r.x = p[0];
  r.y = p[BP];
  return r;
}

// ---------------------------------------------------------------------------
// K1: fused gate GEMMs. All 10 weight tiles share the x/h/c A-tiles in LDS.
// ---------------------------------------------------------------------------
static __device__ __forceinline__ void k1_compute(
    const float* __restrict__ Ax, const float* __restrict__ Ah,
    const float* __restrict__ Ac, const float (*__restrict__ Bs)[KT * BP],
    int wm, int wn, int lane, v8f aF[2], v8f aI[2], v8f aC[2], v8f aO[2]) {
#pragma unroll
  for (int ks = 0; ks < KT; ks += 4) {
    const v2f ax0 = a_frag(Ax, wm * 32, ks, lane);
    const v2f ax1 = a_frag(Ax, wm * 32 + 16, ks, lane);
    const v2f ah0 = a_frag(Ah, wm * 32, ks, lane);
    const v2f ah1 = a_frag(Ah, wm * 32 + 16, ks, lane);
    const v2f ac0 = a_frag(Ac, wm * 32, ks, lane);
    const v2f ac1 = a_frag(Ac, wm * 32 + 16, ks, lane);
    v2f b;
    // forget gate: x, h, c contributions
    b = b_frag(Bs[0], ks, wn * 16, lane);
    aF[0] = wmma_f32x4(ax0, b, aF[0]);
    aF[1] = wmma_f32x4(ax1, b, aF[1]);
    b = b_frag(Bs[4], ks, wn * 16, lane);
    aF[0] = wmma_f32x4(ah0, b, aF[0]);
    aF[1] = wmma_f32x4(ah1, b, aF[1]);
    b = b_frag(Bs[8], ks, wn * 16, lane);
    aF[0] = wmma_f32x4(ac0, b, aF[0]);
    aF[1] = wmma_f32x4(ac1, b, aF[1]);
    // input gate: x, h, c
    b = b_frag(Bs[1], ks, wn * 16, lane);
    aI[0] = wmma_f32x4(ax0, b, aI[0]);
    aI[1] = wmma_f32x4(ax1, b, aI[1]);
    b = b_frag(Bs[5], ks, wn * 16, lane);
    aI[0] = wmma_f32x4(ah0, b, aI[0]);
    aI[1] = wmma_f32x4(ah1, b, aI[1]);
    b = b_frag(Bs[9], ks, wn * 16, lane);
    aI[0] = wmma_f32x4(ac0, b, aI[0]);
    aI[1] = wmma_f32x4(ac1, b, aI[1]);
    // candidate: x, h
    b = b_frag(Bs[2], ks, wn * 16, lane);
    aC[0] = wmma_f32x4(ax0, b, aC[0]);
    aC[1] = wmma_f32x4(ax1, b, aC[1]);
    b = b_frag(Bs[6], ks, wn * 16, lane);
    aC[0] = wmma_f32x4(ah0, b, aC[0]);
    aC[1] = wmma_f32x4(ah1, b, aC[1]);
    // output pre-act (peephole term added in K2): x, h
    b = b_frag(Bs[3], ks, wn * 16, lane);
    aO[0] = wmma_f32x4(ax0, b, aO[0]);
    aO[1] = wmma_f32x4(ax1, b, aO[1]);
    b = b_frag(Bs[7], ks, wn * 16, lane);
    aO[0] = wmma_f32x4(ah0, b, aO[0]);
    aO[1] = wmma_f32x4(ah1, b, aO[1]);
  }
}

__global__ __launch_bounds__(256) void lstm_gates_kernel(
    const float* __restrict__ x, const float* __restrict__ h,
    const float* __restrict__ c, const float* __restrict__ wipf,
    const float* __restrict__ wipi, const float* __restrict__ wipc,
    const float* __restrict__ wipo, const float* __restrict__ wf,
    const float* __restrict__ wi, const float* __restrict__ wc,
    const float* __restrict__ wo, const float* __restrict__ wfc,
    const float* __restrict__ wic, float* __restrict__ Ct_out,
    float* __restrict__ zo_out) {
  const int H = 1024;
  __shared__ float Ax[2][BM * AP];
  __shared__ float Ah[2][BM * AP];
  __shared__ float Ac[2][BM * AP];
  __shared__ float Bs[2][10][KT * BP];

  const int tid = threadIdx.x;
  const int lane = tid & 31;
  const int wave = tid >> 5;
  const int wm = wave & 1;   // 0..1 : 32-row band
  const int wn = wave >> 1;  // 0..3 : 16-col band
  const int col0 = blockIdx.x * BN;
  const int row0 = blockIdx.y * BM;

  v8f aF[2] = {};
  v8f aI[2] = {};
  v8f aC[2] = {};
  v8f aO[2] = {};

  // 13 async b128s per thread per stage.
  auto stage = [&](int buf, int kk) {
    tile_A(Ax[buf], x, row0, kk, H, tid);
    tile_A(Ah[buf], h, row0, kk, H, tid);
    tile_A(Ac[buf], c, row0, kk, H, tid);
    tile_B(Bs[buf][0], wipf, kk, col0, H, tid);
    tile_B(Bs[buf][1], wipi, kk, col0, H, tid);
    tile_B(Bs[buf][2], wipc, kk, col0, H, tid);
    tile_B(Bs[buf][3], wipo, kk, col0, H, tid);
    tile_B(Bs[buf][4], wf, kk, col0, H, tid);
    tile_B(Bs[buf][5], wi, kk, col0, H, tid);
    tile_B(Bs[buf][6], wc, kk, col0, H, tid);
    tile_B(Bs[buf][7], wo, kk, col0, H, tid);
    tile_B(Bs[buf][8], wfc, kk, col0, H, tid);
    tile_B(Bs[buf][9], wic, kk, col0, H, tid);
  };

  stage(0, 0);
  for (int kk = 0; kk < H; kk += KT) {
    const int cur = (kk >> 4) & 1;
    if (kk + KT < H) {
      stage(cur ^ 1, kk + KT);    // prefetch next stage (13 in flight)
      wait_async<13>();           // current stage's 13 copies complete
    } else {
      wait_async<0>();
    }
    __syncthreads();              // LDS visible to whole workgroup
    k1_compute(Ax[cur], Ah[cur], Ac[cur], Bs[cur], wm, wn, lane, aF, aI, aC,
               aO);
    __syncthreads();              // everyone done before cur is overwritten
  }

  // Epilogue: Ct = sigmoid(zf)*c + sigmoid(zi)*tanh(zc); stash zo.
  const int colg = col0 + wn * 16 + (lane & 15);
  const int rbase = row0 + wm * 32 + ((lane >> 4) << 3);
#pragma unroll
  for (int tm = 0; tm < 2; ++tm) {
#pragma unroll
    for (int v = 0; v < 8; ++v) {
      const int row = rbase + tm * 16 + v;
      const size_t idx = (size_t)row * H + colg;
      const float cv = c[idx];
      const float Ft = sigmoidf_(aF[tm][v]);
      const float It = sigmoidf_(aI[tm][v]);
      const float Cd = tanhf(aC[tm][v]);
      Ct_out[idx] = Ft * cv + It * Cd;
      zo_out[idx] = aO[tm][v];
    }
  }
}

// ---------------------------------------------------------------------------
// Generic 64x64-tile GEMM + activation epilogue.
//   MODE 0: out = sigmoid(zo + A@Bw) * tanh(Cin)   (K2, N==H)
//   MODE 1: out = sigmoid(A@Bw)                    (K3)
// ---------------------------------------------------------------------------
template <int MODE>
__global__ __launch_bounds__(256) void gemm_act_kernel(
    const float* __restrict__ A, const float* __restrict__ Bw, int K, int N,
    const float* __restrict__ zo, const float* __restrict__ Cin,
    float* __restrict__ outp) {
  __shared__ float As[2][BM * AP];
  __shared__ float Bsh[2][KT * BP];

  const int tid = threadIdx.x;
  const int lane = tid & 31;
  const int wave = tid >> 5;
  const int wm = wave & 1;
  const int wn = wave >> 1;
  const int col0 = blockIdx.x * BN;
  const int row0 = blockIdx.y * BM;

  v8f acc[2] = {};

  auto stage = [&](int buf, int kk) {
    tile_A(As[buf], A, row0, kk, K, tid);   // 2 async b128s per thread
    tile_B(Bsh[buf], Bw, kk, col0, N, tid);
  };

  stage(0, 0);
  for (int kk = 0; kk < K; kk += KT) {
    const int cur = (kk >> 4) & 1;
    if (kk + KT < K) {
      stage(cur ^ 1, kk + KT);
      wait_async<2>();
    } else {
      wait_async<0>();
    }
    __syncthreads();
#pragma unroll
    for (int ks = 0; ks < KT; ks += 4) {
      const v2f a0 = a_frag(As[cur], wm * 32, ks, lane);
      const v2f a1 = a_frag(As[cur], wm * 32 + 16, ks, lane);
      const v2f b = b_frag(Bsh[cur], ks, wn * 16, lane);
      acc[0] = wmma_f32x4(a0, b, acc[0]);
      acc[1] = wmma_f32x4(a1, b, acc[1]);
    }
    __syncthreads();
  }

  const int colg = col0 + wn * 16 + (lane & 15);
  const int rbase = row0 + wm * 32 + ((lane >> 4) << 3);
#pragma unroll
  for (int tm = 0; tm < 2; ++tm) {
#pragma unroll
    for (int v = 0; v < 8; ++v) {
      const int row = rbase + tm * 16 + v;
      const size_t idx = (size_t)row * N + colg;
      const float zval = acc[tm][v];
      if (MODE == 0) {
        const float Ot = sigmoidf_(zo[idx] + zval);
        outp[idx] = Ot * tanhf(Cin[idx]);
      } else {
        outp[idx] = sigmoidf_(zval);
      }
    }
  }
}

// ---------------------------------------------------------------------------
extern "C" void kernel_launch(void* const* d_in, const int* in_sizes, int n_in,
                              void* d_out, int out_size, void* d_ws,
                              size_t ws_size, hipStream_t stream) {
  (void)in_sizes;
  (void)n_in;
  (void)out_size;
  (void)ws_size;
  const int Bb = 4096, Hh = 1024, Vv = 32000;

  const float* x = (const float*)d_in[0];
  const float* h = (const float*)d_in[1];
  const float* c = (const float*)d_in[2];
  const float* w_forget = (const float*)d_in[3];
  const float* w_input = (const float*)d_in[4];
  const float* w_output = (const float*)d_in[5];
  const float* w_c_dash = (const float*)d_in[6];
  const float* w_forget_c = (const float*)d_in[7];
  const float* w_input_c = (const float*)d_in[8];
  const float* w_output_c = (const float*)d_in[9];
  const float* w_op = (const float*)d_in[10];
  const float* w_ip_f = (const float*)d_in[11];
  const float* w_ip_i = (const float*)d_in[12];
  const float* w_ip_o = (const float*)d_in[13];
  const float* w_ip_cd = (const float*)d_in[14];

  float* Yt = (float*)d_out;
  float* Ht = Yt + (size_t)Bb * Vv;
  float* Ct = Ht + (size_t)Bb * Hh;
  float* zo = (float*)d_ws;  // 4096*1024 floats = 16 MB scratch

  const dim3 blk(256, 1, 1);

  // K1: gates -> Ct, zo
  lstm_gates_kernel<<<dim3(Hh / BN, Bb / BM), blk, 0, stream>>>(
      x, h, c, w_ip_f, w_ip_i, w_ip_cd, w_ip_o, w_forget, w_input, w_c_dash,
      w_output, w_forget_c, w_input_c, Ct, zo);

  // K2: Ht = sigmoid(zo + Ct@w_oc) * tanh(Ct)
  gemm_act_kernel<0><<<dim3(Hh / BN, Bb / BM), blk, 0, stream>>>(
      Ct, w_output_c, Hh, Hh, zo, Ct, Ht);

  // K3: Yt = sigmoid(Ht@w_op); x-fastest grid streams w_op through L2 once
  // per row band (w_op columns for a band = 131 MB total, L2 = 192 MB).
  gemm_act_kernel<1><<<dim3(Vv / BN, Bb / BM), blk, 0, stream>>>(
      Ht, w_op, Hh, Vv, nullptr, nullptr, Yt);
}